// VPC_84018150245085
// MI455X (gfx1250) — compile-verified
//
#include <hip/hip_runtime.h>
#include <math.h>

#define BATCH 8192
#define NFEAT 2048
#define NCLS  10
#define LOG2N 11
#define KTILE 256                       // K-panel staged into LDS per TDM load

typedef __attribute__((ext_vector_type(2))) float    v2f;
typedef __attribute__((ext_vector_type(8))) float    v8f;
typedef __attribute__((ext_vector_type(4))) unsigned v4u;
typedef __attribute__((ext_vector_type(8))) int      v8i;
typedef __attribute__((ext_vector_type(4))) int      v4i;

// ---- workspace layout (in floats) ----
// cw1[2048] | sw1[2048] | BtRe[16*2048] | BtIm[16*2048] | Zre[8192*2048] | Zim[8192*2048]
#define OFF_CW1  0
#define OFF_SW1  (OFF_CW1 + NFEAT)
#define OFF_BTRE (OFF_SW1 + NFEAT)
#define OFF_BTIM (OFF_BTRE + 16 * NFEAT)
#define OFF_ZRE  (OFF_BTIM + 16 * NFEAT)
#define OFF_ZIM  (OFF_ZRE + (size_t)BATCH * NFEAT)

// ---------------------------------------------------------------------------
// Tensor Data Mover: 2D tile (rows x rowlen f32, row stride in f32) -> LDS.
// Descriptor bitfields per CDNA5 ISA.
// ---------------------------------------------------------------------------
#if __has_builtin(__builtin_amdgcn_tensor_load_to_lds)
#define HAVE_TDM 1
__device__ static inline void tdm_load_2d(unsigned lds_off, const float* gptr,
                                          unsigned rows, unsigned rowlen,
                                          unsigned tensor_w, unsigned row_stride) {
    unsigned long long ga = (unsigned long long)(uintptr_t)gptr;
    v4u g0;
    g0[0] = 1u;                                            // count=1, user descriptor
    g0[1] = lds_off;                                       // lds_addr (bytes)
    g0[2] = (unsigned)(ga & 0xffffffffull);                // global_addr[31:0]
    g0[3] = (unsigned)((ga >> 32) & 0x01ffffffull)         // global_addr[56:32]
          | 0x80000000u;                                   // type=2 ("image")
    v8i g1;
    g1[0] = (int)(2u << 16);                               // data_size=2 -> 4 bytes
    g1[1] = (int)((tensor_w & 0xffffu) << 16);             // tensor_dim0[15:0]
    g1[2] = (int)(((tensor_w >> 16) & 0xffffu)             // tensor_dim0[31:16]
          |       ((rows & 0xffffu) << 16));               // tensor_dim1[15:0]
    g1[3] = (int)((rowlen & 0xffffu) << 16);               // tile_dim0
    g1[4] = (int)(rows & 0xffffu);                         // tile_dim1 (tile_dim2=0)
    g1[5] = (int)row_stride;                               // tensor_dim0_stride[31:0]
    g1[6] = 0;                                             // stride hi / dim1_stride lo
    g1[7] = 0;
    v4i gz = {0, 0, 0, 0};                                 // groups 2/3: 2D tensor
#if __has_include(<hip/amd_detail/amd_gfx1250_TDM.h>)
    v8i gz8 = {0, 0, 0, 0, 0, 0, 0, 0};                    // clang-23 6-arg form
    __builtin_amdgcn_tensor_load_to_lds(g0, g1, gz, gz, gz8, 0);
#else
    __builtin_amdgcn_tensor_load_to_lds(g0, g1, gz, gz, 0); // ROCm 7.2 5-arg form
#endif
}
#else
#define HAVE_TDM 0
#endif

// ---------------------------------------------------------------------------
// Kernel 0: per-feature constants: cos/sin(w1), and transposed DFT-16-bin
// coefficient planes Bt[n][k] = e^{-2*pi*i*n*k/N} / sqrt(N)   (n = 0..15)
// ---------------------------------------------------------------------------
__global__ __launch_bounds__(256) void vpc_precompute(const float* __restrict__ weights,
                                                      float* __restrict__ ws) {
    int t = blockIdx.x * blockDim.x + threadIdx.x;   // 0..2047
    if (t >= NFEAT) return;
    float s1, c1;
    __sincosf(weights[NFEAT + t], &s1, &c1);         // layer-1 rotation weights
    ws[OFF_CW1 + t] = c1;
    ws[OFF_SW1 + t] = s1;

    const float invS = 0.022097086912079608f;        // 1/sqrt(2048)
    const float w0   = 6.283185307179586f / (float)NFEAT;
#pragma unroll
    for (int n = 0; n < 16; ++n) {
        int   m = (n * t) & (NFEAT - 1);             // exact periodic reduction
        float sn, cn;
        __sincosf(w0 * (float)m, &sn, &cn);
        ws[OFF_BTRE + n * NFEAT + t] =  cn * invS;
        ws[OFF_BTIM + n * NFEAT + t] = -sn * invS;   // e^{-i theta}
    }
}

// ---------------------------------------------------------------------------
// Kernel 1: one workgroup per batch row.
//   z = exp(i(x+w0)) -> butterfly -> FFT2048 (Stockham DIF, e^{-i}) ->
//   rotate by exp(i w1) -> butterfly -> scale 1/64 -> split re/im planes
// ---------------------------------------------------------------------------
__global__ __launch_bounds__(256) void vpc_layer1(const float* __restrict__ x,
                                                  const float* __restrict__ weights,
                                                  float* __restrict__ ws) {
    __shared__ float sr[2][NFEAT];
    __shared__ float si[2][NFEAT];

    const int   b = blockIdx.x;
    const int   t = threadIdx.x;
    const float inv_sqrt2 = 0.7071067811865476f;
    const float* xrow = x + (size_t)b * NFEAT;

    // ---- layer-1 phase + pairwise butterfly into buffer 0 ----
#pragma unroll
    for (int j = 0; j < 4; ++j) {
        int q  = t + 256 * j;         // pair index 0..1023
        int n0 = 2 * q, n1 = n0 + 1;
        float s0, c0, s1, c1;
        __sincosf(xrow[n0] + weights[n0], &s0, &c0);
        __sincosf(xrow[n1] + weights[n1], &s1, &c1);
        sr[0][n0] = (c0 + c1) * inv_sqrt2;
        si[0][n0] = (s0 + s1) * inv_sqrt2;
        sr[0][n1] = (c0 - c1) * inv_sqrt2;
        si[0][n1] = (s0 - s1) * inv_sqrt2;
    }
    __syncthreads();

    // ---- 11-stage Stockham DIF FFT (forward, e^{-i*theta}) ----
    for (int st = 0; st < LOG2N; ++st) {
        const int   cur = st & 1;
        const int   nxt = cur ^ 1;
        const int   s   = 1 << st;
        const int   n   = NFEAT >> st;
        const float th0 = 6.283185307179586f / (float)n;
#pragma unroll
        for (int j = 0; j < 4; ++j) {
            int i  = t + 256 * j;                    // 0..1023
            int p  = i >> st;
            int o0 = i + (p << st);                  // i + p*s
            float ar = sr[cur][i],        ai = si[cur][i];
            float br = sr[cur][i + 1024], bi = si[cur][i + 1024];
            sr[nxt][o0] = ar + br;
            si[nxt][o0] = ai + bi;
            float dr = ar - br, di = ai - bi;
            float wsn, wcs;
            __sincosf(th0 * (float)p, &wsn, &wcs);   // w = (wcs, -wsn)
            sr[nxt][o0 + s] = dr * wcs + di * wsn;
            si[nxt][o0 + s] = di * wcs - dr * wsn;
        }
        __syncthreads();
    }
    // 11 stages (odd): final data sits in buffer 1

    // ---- layer-2 rotate + butterfly + combined ortho scale, store planes ----
    const float* cw1 = ws + OFF_CW1;
    const float* sw1 = ws + OFF_SW1;
    float* Zre = ws + OFF_ZRE + (size_t)b * NFEAT;
    float* Zim = ws + OFF_ZIM + (size_t)b * NFEAT;
    const float sc = 0.015625f;                      // (1/sqrt(2048)) * (1/sqrt(2)) = 1/64
#pragma unroll
    for (int j = 0; j < 4; ++j) {
        int q  = t + 256 * j;
        int n0 = 2 * q, n1 = n0 + 1;
        float zr0 = sr[1][n0], zi0 = si[1][n0];
        float zr1 = sr[1][n1], zi1 = si[1][n1];
        float r0 = zr0 * cw1[n0] - zi0 * sw1[n0];
        float i0 = zr0 * sw1[n0] + zi0 * cw1[n0];
        float r1 = zr1 * cw1[n1] - zi1 * sw1[n1];
        float i1 = zr1 * sw1[n1] + zi1 * cw1[n1];
        v2f outr, outi;
        outr.x = (r0 + r1) * sc;  outr.y = (r0 - r1) * sc;
        outi.x = (i0 + i1) * sc;  outi.y = (i0 - i1) * sc;
        *(v2f*)(Zre + n0) = outr;
        *(v2f*)(Zim + n0) = outi;
    }
}

// ---------------------------------------------------------------------------
// Kernel 2: 10-bin DFT as complex GEMM via v_wmma_f32_16x16x4_f32.
//   8 waves/block, one wave -> one 16-row x 16-bin tile.
//   B coefficient K-panels (16 x 256, re & im) staged into LDS by the
//   Tensor Data Mover once per block per K-tile; A streamed from global.
//   D_re = A_re*B_re + A_im*(-B_im);  D_im = A_re*B_im + A_im*B_re
//   Epilogue: logits = 5*im/|z|, softmax over 10 classes.
// ---------------------------------------------------------------------------
__global__ __launch_bounds__(256) void vpc_project(const float* __restrict__ ws,
                                                   float* __restrict__ out) {
    __shared__ float sBre[16 * KTILE];               // 16 KB
    __shared__ float sBim[16 * KTILE];               // 16 KB
    __shared__ float tre[8][16][16];                 // 8 KB
    __shared__ float tim[8][16][16];                 // 8 KB

    const int lane = threadIdx.x & 31;
    const int wv   = threadIdx.x >> 5;
    const int tile = blockIdx.x * 8 + wv;            // 0..511
    const int rb   = tile * 16;

    const float* Zre  = ws + OFF_ZRE;
    const float* Zim  = ws + OFF_ZIM;
    const float* BtRe = ws + OFF_BTRE;
    const float* BtIm = ws + OFF_BTIM;

    const int    r16  = lane & 15;
    const int    hi   = lane >> 4;                   // 0: K=0,1  1: K=2,3
    const size_t arow = (size_t)(rb + r16) * NFEAT;  // A fragment row

    const unsigned ldsRe = (unsigned)(uintptr_t)(void*)sBre;  // LDS byte offsets
    const unsigned ldsIm = (unsigned)(uintptr_t)(void*)sBim;

    v8f accRe = {0.f, 0.f, 0.f, 0.f, 0.f, 0.f, 0.f, 0.f};
    v8f accIm = {0.f, 0.f, 0.f, 0.f, 0.f, 0.f, 0.f, 0.f};

    for (int kt = 0; kt < NFEAT / KTILE; ++kt) {
        __syncthreads();                             // previous panel fully consumed
#if HAVE_TDM
        if (threadIdx.x == 0) {                      // wave 0 drives the TDM (EXEC ignored)
            tdm_load_2d(ldsRe, BtRe + kt * KTILE, 16, KTILE, NFEAT, NFEAT);
            tdm_load_2d(ldsIm, BtIm + kt * KTILE, 16, KTILE, NFEAT, NFEAT);
            __builtin_amdgcn_s_wait_tensorcnt(0);    // DMA complete before release
        }
#else
        // Fallback: cooperative staging, 32 floats per thread per plane-pair
        for (int j = threadIdx.x; j < 16 * KTILE / 4; j += 256) {
            int row = j / (KTILE / 4), c4 = j % (KTILE / 4);
            *(v2f*)&sBre[row * KTILE + 4 * c4]     = *(const v2f*)(BtRe + (size_t)row * NFEAT + kt * KTILE + 4 * c4);
            *(v2f*)&sBre[row * KTILE + 4 * c4 + 2] = *(const v2f*)(BtRe + (size_t)row * NFEAT + kt * KTILE + 4 * c4 + 2);
            *(v2f*)&sBim[row * KTILE + 4 * c4]     = *(const v2f*)(BtIm + (size_t)row * NFEAT + kt * KTILE + 4 * c4);
            *(v2f*)&sBim[row * KTILE + 4 * c4 + 2] = *(const v2f*)(BtIm + (size_t)row * NFEAT + kt * KTILE + 4 * c4 + 2);
        }
#endif
        __syncthreads();                             // panel visible to all 8 waves

        for (int c = 0; c < KTILE / 4; ++c) {
            const int kk = 4 * c + 2 * hi;           // K offset within panel
            const int k  = kt * KTILE + kk;          // global K
            if (c == 0 && kt + 1 < NFEAT / KTILE) {  // uniform branch; EXEC unchanged
                __builtin_prefetch(Zre + arow + k + KTILE, 0, 1);
                __builtin_prefetch(Zim + arow + k + KTILE, 0, 1);
            }
            v2f aR = *(const v2f*)(Zre + arow + k);
            v2f aI = *(const v2f*)(Zim + arow + k);
            v2f bR = *(const v2f*)&sBre[r16 * KTILE + kk];
            v2f bI = *(const v2f*)&sBim[r16 * KTILE + kk];
            v2f nbI = -bI;
            accRe = __builtin_amdgcn_wmma_f32_16x16x4_f32(false, aR, false, bR,  (short)0, accRe, false, false);
            accRe = __builtin_amdgcn_wmma_f32_16x16x4_f32(false, aI, false, nbI, (short)0, accRe, false, false);
            accIm = __builtin_amdgcn_wmma_f32_16x16x4_f32(false, aR, false, bI,  (short)0, accIm, false, false);
            accIm = __builtin_amdgcn_wmma_f32_16x16x4_f32(false, aI, false, bR,  (short)0, accIm, false, false);
        }
    }

    // scatter C/D tile (VGPR r: lanes 0-15 -> M=r, lanes 16-31 -> M=8+r)
#pragma unroll
    for (int r = 0; r < 8; ++r) {
        tre[wv][hi * 8 + r][r16] = accRe[r];
        tim[wv][hi * 8 + r][r16] = accIm[r];
    }
    __syncthreads();

    if (lane < 16) {                                 // one thread per row of the tile
        float lg[NCLS];
        float mx = -1e30f;
#pragma unroll
        for (int n = 0; n < NCLS; ++n) {
            float re = tre[wv][lane][n];
            float im = tim[wv][lane][n];
            float l  = 5.0f * im * rsqrtf(fmaxf(re * re + im * im, 1e-30f)); // 5*sin(atan2)
            lg[n] = l;
            mx = fmaxf(mx, l);
        }
        float sum = 0.f;
#pragma unroll
        for (int n = 0; n < NCLS; ++n) { float e = __expf(lg[n] - mx); lg[n] = e; sum += e; }
        const float inv = 1.0f / sum;
        float* orow = out + (size_t)(rb + lane) * NCLS;
#pragma unroll
        for (int n = 0; n < NCLS; ++n) orow[n] = lg[n] * inv;
    }
}

// ---------------------------------------------------------------------------
extern "C" void kernel_launch(void* const* d_in, const int* in_sizes, int n_in,
                              void* d_out, int out_size, void* d_ws, size_t ws_size,
                              hipStream_t stream) {
    const float* x = (const float*)d_in[0];   // [8192, 2048]
    const float* w = (const float*)d_in[1];   // [2*2048]
    float* out = (float*)d_out;               // [8192, 10]
    float* ws  = (float*)d_ws;                // needs ~134.5 MB

    vpc_precompute<<<NFEAT / 256, 256, 0, stream>>>(w, ws);
    vpc_layer1<<<BATCH, 256, 0, stream>>>(x, w, ws);
    vpc_project<<<BATCH / (16 * 8), 256, 0, stream>>>(ws, out);
}